// GCNModel_64390149701818
// MI455X (gfx1250) — compile-verified
//
#include <hip/hip_runtime.h>
#include <hip/hip_bf16.h>
#include <math.h>

typedef __attribute__((ext_vector_type(2))) float v2f;
typedef __attribute__((ext_vector_type(8))) float v8f;

#define HIDF 384
#define EPSF 1e-5f

// GEMM tiling
#define BM 128          // rows per block (8 waves x 16)
#define BN 64           // cols per block (4 col-tiles per wave)
#define KB 32           // k-depth per stage (8 wmma steps)
#define APAD 36         // LDS row stride for A (bank-conflict-free, 8B aligned)

__device__ __forceinline__ float warpSum(float v) {
#pragma unroll
  for (int o = 16; o > 0; o >>= 1) v += __shfl_xor(v, o, 32);
  return v;
}

__device__ __forceinline__ void atomicMaxF(float* addr, float v) {
  if (v >= 0.0f) atomicMax((int*)addr, __float_as_int(v));
  else           atomicMin((unsigned int*)addr, __float_as_uint(v));
}

// ================= WMMA GEMM: C[Nr,M] = A[Nr,K] @ W[K,M] (+bias) ==========
// Block: 8 waves -> 128x64 tile. A staged 128xKB in LDS (padded), B staged
// pre-packed in WMMA fragment order (float2 per lane per k-step).
__global__ __launch_bounds__(256)
void wmma_gemm_f32(const float* __restrict__ A, const float* __restrict__ W,
                   const float* __restrict__ bias, float* __restrict__ C,
                   int Nr, int K, int M) {
  __shared__ float  lA[BM * APAD];                        // 18 KB
  __shared__ float2 lB[(KB / 4) * (BN / 16) * 32];        // 8 KB

  const int tid  = threadIdx.x;
  const int lane = tid & 31;
  const int wave = tid >> 5;

  const int nColBlk = M / BN;
  const int rowBase = (blockIdx.x / nColBlk) * BM;
  const int colBase = (blockIdx.x % nColBlk) * BN;

  const int hlf  = lane >> 4;       // 0/1
  const int l15  = lane & 15;
  const int koff = hlf * 2;

  const int  myRow0 = rowBase + wave * 16;
  const bool active = (myRow0 < Nr);      // wave-uniform

  v8f acc0 = {}, acc1 = {}, acc2 = {}, acc3 = {};

  // A staging: thread -> (row = tid>>1, half-row of 16 floats)
  const int sRow  = tid >> 1;
  const int sHalf = tid & 1;
  int gRow = rowBase + sRow; if (gRow >= Nr) gRow = Nr - 1;
  const float* gA = A + (size_t)gRow * K + sHalf * 16;
  float*       sA = lA + sRow * APAD + sHalf * 16;

  for (int k0 = 0; k0 < K; k0 += KB) {
    // ---- stage A: BM x KB (coalesced b128 loads) ----
    {
      const float* p = gA + k0;
#pragma unroll
      for (int i = 0; i < 4; ++i)
        *(float4*)(sA + i * 4) = *(const float4*)(p + i * 4);
      if (k0 + KB < K) __builtin_prefetch(p + KB, 0, 3);   // global_prefetch_b8
    }
    // ---- stage B packed in fragment order: 1024 float2 entries, 4/thread ----
#pragma unroll
    for (int i = 0; i < 4; ++i) {
      int e    = tid + 256 * i;
      int kk   = e >> 7;            // /(4*32)
      int rem  = e & 127;
      int ct   = rem >> 5;
      int ln   = rem & 31;
      int bcol = colBase + ct * 16 + (ln & 15);
      int krow = k0 + kk * 4 + (ln >> 4) * 2;
      float2 t;
      t.x = W[(size_t)krow * M + bcol];
      t.y = W[(size_t)(krow + 1) * M + bcol];
      lB[e] = t;
    }
    __syncthreads();

    if (active) {
      const float* aRow = lA + (wave * 16 + l15) * APAD;
#pragma unroll
      for (int kk = 0; kk < KB / 4; ++kk) {
        float2 av = *(const float2*)(aRow + kk * 4 + koff);   // ds_load_b64
        v2f a; a.x = av.x; a.y = av.y;
        const float2* bb = lB + kk * 128 + lane;
        float2 b0 = bb[0], b1 = bb[32], b2 = bb[64], b3 = bb[96];
        v2f vb;
        vb.x = b0.x; vb.y = b0.y;
        acc0 = __builtin_amdgcn_wmma_f32_16x16x4_f32(false, a, false, vb, (short)0, acc0, false, false);
        vb.x = b1.x; vb.y = b1.y;
        acc1 = __builtin_amdgcn_wmma_f32_16x16x4_f32(false, a, false, vb, (short)0, acc1, false, false);
        vb.x = b2.x; vb.y = b2.y;
        acc2 = __builtin_amdgcn_wmma_f32_16x16x4_f32(false, a, false, vb, (short)0, acc2, false, false);
        vb.x = b3.x; vb.y = b3.y;
        acc3 = __builtin_amdgcn_wmma_f32_16x16x4_f32(false, a, false, vb, (short)0, acc3, false, false);
      }
    }
    __syncthreads();
  }

  if (active) {
    const int crow = myRow0 + hlf * 8;
#pragma unroll
    for (int ct = 0; ct < 4; ++ct) {
      const v8f acc = ct == 0 ? acc0 : ct == 1 ? acc1 : ct == 2 ? acc2 : acc3;
      const int ccol = colBase + ct * 16 + l15;
      const float bv = bias ? bias[ccol] : 0.0f;
#pragma unroll
      for (int r = 0; r < 8; ++r)
        C[(size_t)(crow + r) * M + ccol] = acc[r] + bv;
    }
  }
}

// ============================ small utilities =============================
__global__ void k_fill(float* p, float v, int n) {
  int i = blockIdx.x * blockDim.x + threadIdx.x;
  if (i < n) p[i] = v;
}

__global__ void k_deg(const int* __restrict__ dst, float* deg, int E) {
  int i = blockIdx.x * blockDim.x + threadIdx.x;
  if (i < E) atomicAdd(&deg[dst[i]], 1.0f);
}

__global__ void k_dinv(float* d, int N) {
  int i = blockIdx.x * blockDim.x + threadIdx.x;
  if (i < N) d[i] = rsqrtf(d[i] + 1.0f);
}

// =============================== GCN ======================================
__global__ void k_gcn_scatter(const int* __restrict__ src, const int* __restrict__ dst,
                              const float* __restrict__ dinv, const float* __restrict__ xw,
                              float* __restrict__ out, int E) {
  int w = (blockIdx.x * blockDim.x + threadIdx.x) >> 5;
  int lane = threadIdx.x & 31;
  if (w >= E) return;
  int s = src[w], d = dst[w];
  float coef = dinv[s] * dinv[d];
  const float* r = xw + (size_t)s * HIDF;
  float* o = out + (size_t)d * HIDF;
  for (int c = lane; c < HIDF; c += 32) atomicAdd(&o[c], coef * r[c]);
}

__global__ void k_gcn_self(const float* __restrict__ xw, const float* __restrict__ dinv,
                           float* __restrict__ out, int N) {
  int i = blockIdx.x * blockDim.x + threadIdx.x;
  if (i >= N * HIDF) return;
  int n = i / HIDF;
  float di = dinv[n];
  out[i] += xw[i] * di * di;
}

// =============================== GAT ======================================
__global__ void k_gat_att(const float* __restrict__ xw, const float* __restrict__ as_,
                          const float* __restrict__ ad_, float* __restrict__ asrc,
                          float* __restrict__ adst, int N) {
  int w = (blockIdx.x * blockDim.x + threadIdx.x) >> 5;
  int lane = threadIdx.x & 31;
  if (w >= N) return;
  const float* row = xw + (size_t)w * (2 * HIDF);
  float s0 = 0, s1 = 0, d0 = 0, d1 = 0;
  for (int c = lane; c < HIDF; c += 32) {
    float x0 = row[c], x1 = row[HIDF + c];
    s0 += x0 * as_[c];        s1 += x1 * as_[HIDF + c];
    d0 += x0 * ad_[c];        d1 += x1 * ad_[HIDF + c];
  }
  s0 = warpSum(s0); s1 = warpSum(s1); d0 = warpSum(d0); d1 = warpSum(d1);
  if (lane == 0) {
    asrc[2 * w] = s0; asrc[2 * w + 1] = s1;
    adst[2 * w] = d0; adst[2 * w + 1] = d1;
  }
}

__device__ __forceinline__ float lrelu(float e) { return e > 0.0f ? e : 0.2f * e; }

__global__ void k_gat_max(const int* src, const int* dst, const float* asrc,
                          const float* adst, float* m, int E, int N) {
  int t = blockIdx.x * blockDim.x + threadIdx.x;
  if (t >= (E + N) * 2) return;
  int idx = t >> 1, h = t & 1;
  int s = idx < E ? src[idx] : idx - E;
  int d = idx < E ? dst[idx] : idx - E;
  atomicMaxF(&m[2 * d + h], lrelu(asrc[2 * s + h] + adst[2 * d + h]));
}

__global__ void k_gat_denom(const int* src, const int* dst, const float* asrc,
                            const float* adst, const float* m, float* dnm,
                            int E, int N) {
  int t = blockIdx.x * blockDim.x + threadIdx.x;
  if (t >= (E + N) * 2) return;
  int idx = t >> 1, h = t & 1;
  int s = idx < E ? src[idx] : idx - E;
  int d = idx < E ? dst[idx] : idx - E;
  float e = lrelu(asrc[2 * s + h] + adst[2 * d + h]);
  atomicAdd(&dnm[2 * d + h], expf(e - m[2 * d + h]));
}

__global__ void k_gat_msg(const int* src, const int* dst, const float* asrc,
                          const float* adst, const float* m, const float* dnm,
                          const float* __restrict__ xw, float* __restrict__ out,
                          int E, int N) {
  int w = (blockIdx.x * blockDim.x + threadIdx.x) >> 5;
  int lane = threadIdx.x & 31;
  if (w >= E + N) return;
  int s = w < E ? src[w] : w - E;
  int d = w < E ? dst[w] : w - E;
  float e0 = lrelu(asrc[2 * s] + adst[2 * d]);
  float e1 = lrelu(asrc[2 * s + 1] + adst[2 * d + 1]);
  float a0 = expf(e0 - m[2 * d]) / dnm[2 * d];
  float a1 = expf(e1 - m[2 * d + 1]) / dnm[2 * d + 1];
  const float* r0 = xw + (size_t)s * (2 * HIDF);
  const float* r1 = r0 + HIDF;
  float* o = out + (size_t)d * HIDF;
  for (int c = lane; c < HIDF; c += 32)
    atomicAdd(&o[c], 0.5f * (a0 * r0[c] + a1 * r1[c]));
}

// ======================== BatchNorm (column stats) ========================
__global__ void k_colstats(const float* __restrict__ X, int Nr, int M,
                           float* sum, float* sq) {
  int c = threadIdx.x;
  if (c >= M) return;
  int rpb = (Nr + gridDim.x - 1) / gridDim.x;
  int r0 = blockIdx.x * rpb;
  int r1 = r0 + rpb; if (r1 > Nr) r1 = Nr;
  float s = 0, q = 0;
  for (int r = r0; r < r1; ++r) {
    float v = X[(size_t)r * M + c];
    s += v; q += v * v;
  }
  atomicAdd(&sum[c], s);
  atomicAdd(&sq[c], q);
}

__global__ void k_bn_apply(float* __restrict__ X, const float* __restrict__ prev,
                           const float* __restrict__ g, const float* __restrict__ b,
                           const float* sum, const float* sq, int Nr, int M,
                           int residual) {
  int i = blockIdx.x * blockDim.x + threadIdx.x;
  if (i >= Nr * M) return;
  int c = i % M;
  float invN = 1.0f / (float)Nr;
  float mu = sum[c] * invN;
  float var = sq[c] * invN - mu * mu;
  float y = (X[i] - mu) * rsqrtf(var + EPSF) * g[c] + b[c];
  y = y > 0.0f ? y : 0.0f;
  if (residual) y += prev[i];
  X[i] = y;
}

// ============================== pooling ===================================
__global__ void k_count(const int* batch, float* cnt, int N) {
  int i = blockIdx.x * blockDim.x + threadIdx.x;
  if (i < N) atomicAdd(&cnt[batch[i]], 1.0f);
}

__global__ void k_pool_init(float* z, int G) {
  int i = blockIdx.x * blockDim.x + threadIdx.x;
  if (i >= G * HIDF) return;
  z[(size_t)(i / HIDF) * (2 * HIDF) + HIDF + (i % HIDF)] = -INFINITY;
}

__global__ void k_pool(const float* __restrict__ h, const int* batch,
                       float* __restrict__ z, int N) {
  int w = (blockIdx.x * blockDim.x + threadIdx.x) >> 5;
  int lane = threadIdx.x & 31;
  if (w >= N) return;
  int g = batch[w];
  const float* r = h + (size_t)w * HIDF;
  float* zm = z + (size_t)g * (2 * HIDF);
  for (int c = lane; c < HIDF; c += 32) {
    float v = r[c];
    atomicAdd(&zm[c], v);
    atomicMaxF(&zm[HIDF + c], v);
  }
}

__global__ void k_pool_fin(float* z, const float* cnt, int G) {
  int i = blockIdx.x * blockDim.x + threadIdx.x;
  if (i >= G * 2 * HIDF) return;
  int g = i / (2 * HIDF), c = i % (2 * HIDF);
  if (c < HIDF) {
    float n = cnt[g]; if (n < 1.0f) n = 1.0f;
    z[i] /= n;
  } else {
    float v = z[i];
    if (!isfinite(v)) z[i] = 0.0f;
  }
}

// ====================== classifier tail kernels ===========================
__global__ void k_ln_relu(float* __restrict__ z2, const float* g, const float* b,
                          int G) {
  int w = (blockIdx.x * blockDim.x + threadIdx.x) >> 5;
  int lane = threadIdx.x & 31;
  if (w >= G) return;
  float* r = z2 + (size_t)w * 192;
  float s = 0, q = 0;
  for (int c = lane; c < 192; c += 32) { float v = r[c]; s += v; q += v * v; }
  s = warpSum(s); q = warpSum(q);
  float mu = s / 192.0f;
  float var = q / 192.0f - mu * mu;
  float inv = rsqrtf(var + EPSF);
  for (int c = lane; c < 192; c += 32) {
    float v = (r[c] - mu) * inv * g[c] + b[c];
    r[c] = v > 0.0f ? v : 0.0f;
  }
}

__global__ void k_final(const float* __restrict__ z2, const float* __restrict__ w3,
                        const float* __restrict__ b3, float* __restrict__ out, int G) {
  int t = blockIdx.x * blockDim.x + threadIdx.x;
  if (t >= G * 2) return;
  int g = t >> 1, c = t & 1;
  const float* r = z2 + (size_t)g * 192;
  float s = 0;
  for (int k = 0; k < 192; ++k) s += r[k] * w3[2 * k + c];
  out[t] = s + b3[c];
}

// =============================== host =====================================
extern "C" void kernel_launch(void* const* d_in, const int* in_sizes, int n_in,
                              void* d_out, int out_size, void* d_ws, size_t ws_size,
                              hipStream_t stream) {
  const int N = in_sizes[0] / 128;
  const int E = in_sizes[1] / 2;
  const int G = out_size / 2;

  const float* x      = (const float*)d_in[0];
  const int*   ei     = (const int*)d_in[1];
  const int*   src    = ei;
  const int*   dst    = ei + E;
  const int*   batch  = (const int*)d_in[2];
  const float* gcn0_w = (const float*)d_in[3];
  const float* gat1_w = (const float*)d_in[5];
  const float* gat1_as = (const float*)d_in[6];
  const float* gat1_ad = (const float*)d_in[7];
  const float* gcn2_w = (const float*)d_in[9];
  const float* gat3_w = (const float*)d_in[11];
  const float* gat3_as = (const float*)d_in[12];
  const float* gat3_ad = (const float*)d_in[13];
  const float* gcn4_w = (const float*)d_in[15];
  const float* bn_g   = (const float*)d_in[17];
  const float* bn_b   = (const float*)d_in[18];
  const float* clf_w1 = (const float*)d_in[19];
  const float* clf_b1 = (const float*)d_in[20];
  const float* clf_bng = (const float*)d_in[21];
  const float* clf_bnb = (const float*)d_in[22];
  const float* clf_w2 = (const float*)d_in[23];
  const float* clf_b2 = (const float*)d_in[24];
  const float* clf_lng = (const float*)d_in[25];
  const float* clf_lnb = (const float*)d_in[26];
  const float* clf_w3 = (const float*)d_in[27];
  const float* clf_b3 = (const float*)d_in[28];
  float* out = (float*)d_out;

  // workspace carve-up
  float* W0 = (float*)d_ws;
  size_t off = 0;
  auto alloc = [&](size_t n) { float* p = W0 + off; off += n; return p; };
  float* xw   = alloc((size_t)N * 768);
  float* bufA = alloc((size_t)N * HIDF);
  float* bufB = alloc((size_t)N * HIDF);
  float* dinv = alloc(N);
  float* asrc = alloc((size_t)N * 2);
  float* adst = alloc((size_t)N * 2);
  float* mbuf = alloc((size_t)N * 2);
  float* dnm  = alloc((size_t)N * 2);
  float* stats = alloc(768);
  float* cnts = alloc(G);
  float* z    = alloc((size_t)G * 768);
  float* z1   = alloc((size_t)G * HIDF);
  float* z2   = alloc((size_t)G * 192);

  auto cdiv = [](long long a, long long b) { return (int)((a + b - 1) / b); };

  auto gemm = [&](const float* A, const float* Wt, const float* bias, float* C,
                  int Nr, int K, int M) {
    int blocks = cdiv(Nr, BM) * (M / BN);
    wmma_gemm_f32<<<blocks, 256, 0, stream>>>(A, Wt, bias, C, Nr, K, M);
  };

  // degree / symmetric normalization (once, reused by all GCN layers)
  hipMemsetAsync(dinv, 0, (size_t)N * sizeof(float), stream);
  k_deg<<<cdiv(E, 256), 256, 0, stream>>>(dst, dinv, E);
  k_dinv<<<cdiv(N, 256), 256, 0, stream>>>(dinv, N);

  auto gcn = [&](const float* h, int K, const float* w, float* o) {
    hipMemsetAsync(o, 0, (size_t)N * HIDF * sizeof(float), stream);
    gemm(h, w, nullptr, xw, N, K, HIDF);
    k_gcn_scatter<<<cdiv((long long)E * 32, 256), 256, 0, stream>>>(src, dst, dinv, xw, o, E);
    k_gcn_self<<<cdiv((long long)N * HIDF, 256), 256, 0, stream>>>(xw, dinv, o, N);
  };

  auto gat = [&](const float* h, const float* w, const float* as_, const float* ad_,
                 float* o) {
    gemm(h, w, nullptr, xw, N, HIDF, 2 * HIDF);
    k_gat_att<<<cdiv((long long)N * 32, 256), 256, 0, stream>>>(xw, as_, ad_, asrc, adst, N);
    k_fill<<<cdiv(N * 2, 256), 256, 0, stream>>>(mbuf, -INFINITY, N * 2);
    hipMemsetAsync(dnm, 0, (size_t)N * 2 * sizeof(float), stream);
    hipMemsetAsync(o, 0, (size_t)N * HIDF * sizeof(float), stream);
    int tot2 = (E + N) * 2;
    k_gat_max<<<cdiv(tot2, 256), 256, 0, stream>>>(src, dst, asrc, adst, mbuf, E, N);
    k_gat_denom<<<cdiv(tot2, 256), 256, 0, stream>>>(src, dst, asrc, adst, mbuf, dnm, E, N);
    k_gat_msg<<<cdiv((long long)(E + N) * 32, 256), 256, 0, stream>>>(src, dst, asrc, adst,
                                                                      mbuf, dnm, xw, o, E, N);
  };

  auto bn = [&](float* X, const float* prev, const float* g, const float* b, int Nr) {
    hipMemsetAsync(stats, 0, 768 * sizeof(float), stream);
    int nb = Nr >= 4096 ? 128 : 8;
    k_colstats<<<nb, HIDF, 0, stream>>>(X, Nr, HIDF, stats, stats + HIDF);
    k_bn_apply<<<cdiv((long long)Nr * HIDF, 256), 256, 0, stream>>>(
        X, prev, g, b, stats, stats + HIDF, Nr, HIDF, prev != nullptr);
  };

  // ----- 5 conv layers, ping-pong bufA/bufB -----
  gcn(x, 128, gcn0_w, bufA);                 bn(bufA, nullptr, bn_g + 0*HIDF, bn_b + 0*HIDF, N);
  gat(bufA, gat1_w, gat1_as, gat1_ad, bufB); bn(bufB, bufA,   bn_g + 1*HIDF, bn_b + 1*HIDF, N);
  gcn(bufB, HIDF, gcn2_w, bufA);             bn(bufA, bufB,   bn_g + 2*HIDF, bn_b + 2*HIDF, N);
  gat(bufA, gat3_w, gat3_as, gat3_ad, bufB); bn(bufB, bufA,   bn_g + 3*HIDF, bn_b + 3*HIDF, N);
  gcn(bufB, HIDF, gcn4_w, bufA);             bn(bufA, bufB,   bn_g + 4*HIDF, bn_b + 4*HIDF, N);

  // ----- global mean/max pool -> z[G, 768] -----
  hipMemsetAsync(cnts, 0, (size_t)G * sizeof(float), stream);
  hipMemsetAsync(z, 0, (size_t)G * 768 * sizeof(float), stream);
  k_pool_init<<<cdiv((long long)G * HIDF, 256), 256, 0, stream>>>(z, G);
  k_count<<<cdiv(N, 256), 256, 0, stream>>>(batch, cnts, N);
  k_pool<<<cdiv((long long)N * 32, 256), 256, 0, stream>>>(bufA, batch, z, N);
  k_pool_fin<<<cdiv((long long)G * 768, 256), 256, 0, stream>>>(z, cnts, G);

  // ----- classifier -----
  gemm(z, clf_w1, clf_b1, z1, G, 768, HIDF);
  bn(z1, nullptr, clf_bng, clf_bnb, G);
  gemm(z1, clf_w2, clf_b2, z2, G, HIDF, 192);
  k_ln_relu<<<cdiv((long long)G * 32, 256), 256, 0, stream>>>(z2, clf_lng, clf_lnb, G);
  k_final<<<cdiv(G * 2, 256), 256, 0, stream>>>(z2, clf_w3, clf_b3, out, G);
}